// GNNDeleteModel_89670327206041
// MI455X (gfx1250) — compile-verified
//
#include <hip/hip_runtime.h>
#include <hip/hip_bf16.h>

// GNNDeleteModel for MI455X (gfx1250).
// Pipeline:
//   deg/dinv -> HW1 = x@Wc1 (WMMA f32) -> agg1 = self + edge scatter + bc1
//   -> relu -> HL = h@Wd1^T (WMMA) -> h_del = mask ? HL+bd1 : h
//   -> HW2 = h_del@Wc2 (WMMA) -> agg2 = self + edge scatter + bc2
//   -> OL = agg2@Wd2^T (WMMA) -> out = mask ? OL+bd2 : agg2
// All GEMM shapes are compile-time constants so every A/B access lowers to
// global_load with an immediate offset and the K loop fully unrolls into a
// straight run of v_wmma_f32_16x16x4_f32.

typedef __attribute__((ext_vector_type(2))) float v2f;
typedef __attribute__((ext_vector_type(8))) float v8f;

#define N_IN   128
#define N_HID  64
#define N_OUT  40

// ---------------------------------------------------------------------------
// Degree / normalization kernels
// ---------------------------------------------------------------------------
__global__ void deg_init_kernel(float* __restrict__ deg, int n) {
    int i = blockIdx.x * blockDim.x + threadIdx.x;
    if (i < n) deg[i] = 1.0f;  // self-loop contributes 1 to every node's degree
}

__global__ void deg_accum_kernel(const int* __restrict__ dst,
                                 float* __restrict__ deg, int e) {
    int i = blockIdx.x * blockDim.x + threadIdx.x;
    if (i < e) atomicAdd(&deg[dst[i]], 1.0f);
}

__global__ void dinv_kernel(float* __restrict__ deg, int n) {
    int i = blockIdx.x * blockDim.x + threadIdx.x;
    if (i < n) {
        float d = deg[i];
        deg[i] = (d > 0.0f) ? rsqrtf(d) : 0.0f;  // in-place: deg -> dinv
    }
}

// ---------------------------------------------------------------------------
// Generic fp32 WMMA GEMM:  O[M x NCOLS] = A[M x K] * B  (all shapes static)
//   TRANSB == 0 : B is K x NCOLS, row stride LDB       (B[k][c] = B[k*LDB+c])
//   TRANSB == 1 : B is NCOLS x K, row stride LDB (B^T) (B[k][c] = B[c*LDB+k])
// One wave computes one 16x16 output tile via V_WMMA_F32_16X16X4_F32.
// M % 16 == 0, K % 4 == 0. NCOLS may be ragged: the B-column index is
// CLAMPED in-bounds (column n of B only influences column n of D, so
// out-of-range lanes compute garbage in their own column only) and the
// store is guarded. EXEC stays all-1s for every WMMA.
// blockDim.x = 128 (4 waves); grid = (ceil(mtiles/4), ceil(NCOLS/16)).
// ---------------------------------------------------------------------------
template <int LDA, int LDB, int TRANSB, int K, int NCOLS, int LDO>
__global__ __launch_bounds__(128) void wmma_gemm_f32(
    const float* __restrict__ A, const float* __restrict__ B,
    float* __restrict__ O, int mtiles)
{
    const int lane = threadIdx.x & 31;
    const int wave = threadIdx.x >> 5;
    const int rt   = blockIdx.x * (blockDim.x >> 5) + wave;  // row tile (wave-uniform)
    const int nt   = blockIdx.y;                             // col tile
    if (rt >= mtiles) return;                                // uniform per wave

    const int m    = lane & 15;   // 0..15
    const int half = lane >> 4;   // 0..1 : which K-pair this lane holds
    const int row  = rt * 16 + m;
    const int col  = nt * 16 + m;
    const int colc = (NCOLS % 16 == 0) ? col
                   : ((col < NCOLS) ? col : (NCOLS - 1));    // clamped (legal loads)

    const float* __restrict__ aptr = A + (size_t)row * LDA;
    const float* __restrict__ bptr = TRANSB ? (B + (size_t)colc * LDB) : (B + colc);
    constexpr int BK = TRANSB ? 1 : LDB;                     // B stride per K step

    v8f acc = {};
    #pragma unroll
    for (int k = 0; k < K; k += 4) {
        const int ka = k + 2 * half;  // lanes 0-15 hold K=k,k+1 ; lanes 16-31 K=k+2,k+3
        v2f a, b;
        a.x = aptr[ka];
        a.y = aptr[ka + 1];
        b.x = bptr[ka * BK];
        b.y = bptr[(ka + 1) * BK];
        // D = A(16x4,f32) * B(4x16,f32) + C(16x16,f32)
        acc = __builtin_amdgcn_wmma_f32_16x16x4_f32(
            /*neg_a=*/false, a, /*neg_b=*/false, b,
            /*c_mod=*/(short)0, acc, /*reuse_a=*/false, /*reuse_b=*/false);
    }

    if (NCOLS % 16 == 0 || col < NCOLS) {
        // C/D layout: VGPR r -> row (r + 8*half) of the 16x16 tile, col = m
        const int orow = rt * 16 + half * 8;
        #pragma unroll
        for (int r = 0; r < 8; ++r)
            O[(size_t)(orow + r) * LDO + col] = acc[r];
    }
}

// ---------------------------------------------------------------------------
// agg[i][c] = dinv[i]^2 * hw[i][c] + bias[c]   (self-loop message + bias)
// ---------------------------------------------------------------------------
template <int C>
__global__ void agg_init_kernel(const float* __restrict__ hw,
                                const float* __restrict__ dinv,
                                const float* __restrict__ bias,
                                float* __restrict__ agg, int n) {
    int idx = blockIdx.x * blockDim.x + threadIdx.x;
    if (idx >= n * C) return;
    int i = idx / C;
    int c = idx - i * C;
    float di = dinv[i];
    agg[idx] = di * di * hw[idx] + bias[c];
}

// ---------------------------------------------------------------------------
// Edge scatter: agg[dst] += dinv[src]*dinv[dst] * hw[src]  (one thread/edge)
// ---------------------------------------------------------------------------
template <int C>
__global__ void edge_scatter_kernel(const int* __restrict__ src,
                                    const int* __restrict__ dst,
                                    const float* __restrict__ dinv,
                                    const float* __restrict__ hw,
                                    float* __restrict__ agg, int e) {
    int i = blockIdx.x * blockDim.x + threadIdx.x;
    if (i >= e) return;
    int s = src[i];
    int d = dst[i];
    float nrm = dinv[s] * dinv[d];
    const float* hs = hw + (size_t)s * C;
    float* ad = agg + (size_t)d * C;
    #pragma unroll
    for (int c = 0; c < C; c += 4) {
        float4 v = *(const float4*)(hs + c);
        atomicAdd(ad + c + 0, nrm * v.x);
        atomicAdd(ad + c + 1, nrm * v.y);
        atomicAdd(ad + c + 2, nrm * v.z);
        atomicAdd(ad + c + 3, nrm * v.w);
    }
}

__global__ void relu_kernel(float* __restrict__ a, int n) {
    int i = blockIdx.x * blockDim.x + threadIdx.x;
    if (i < n) a[i] = fmaxf(a[i], 0.0f);
}

// out[i][c] = mask[i] ? lin[i][c] + bias[c] : base[i][c]   (out may alias lin)
template <int C>
__global__ void select_affected_kernel(const float* __restrict__ lin,
                                       const float* __restrict__ bias,
                                       const float* __restrict__ base,
                                       const unsigned char* __restrict__ mask,
                                       float* __restrict__ out, int n) {
    int idx = blockIdx.x * blockDim.x + threadIdx.x;
    if (idx >= n * C) return;
    int i = idx / C;
    int c = idx - i * C;
    float lv = lin[idx] + bias[c];
    float bv = base[idx];
    out[idx] = mask[i] ? lv : bv;
}

static inline int cdiv(int a, int b) { return (a + b - 1) / b; }

extern "C" void kernel_launch(void* const* d_in, const int* in_sizes, int n_in,
                              void* d_out, int out_size, void* d_ws, size_t ws_size,
                              hipStream_t stream) {
    const float*          x    = (const float*)d_in[0];
    const int*            ei   = (const int*)d_in[1];
    const unsigned char*  mask = (const unsigned char*)d_in[2];
    const float* Wc1 = (const float*)d_in[3];
    const float* bc1 = (const float*)d_in[4];
    const float* Wc2 = (const float*)d_in[5];
    const float* bc2 = (const float*)d_in[6];
    const float* Wd1 = (const float*)d_in[7];
    const float* bd1 = (const float*)d_in[8];
    const float* Wd2 = (const float*)d_in[9];
    const float* bd2 = (const float*)d_in[10];

    const int N = in_sizes[2];          // affected_mask has N elements
    const int E = in_sizes[1] / 2;      // edge_index is (2, E)
    const int* srcIdx = ei;             // edge_index[0]
    const int* dstIdx = ei + E;         // edge_index[1]

    // Workspace carve-up (floats):
    //   dinv : N
    //   B1   : N*64   HW1 -> HL -> h_del
    //   B2   : N*64   agg1 -> h (relu in place)
    //   B3   : N*40   HW2 -> OL
    //   B4   : N*40   agg2
    float* dinv = (float*)d_ws;
    float* B1 = dinv + N;
    float* B2 = B1 + (size_t)N * N_HID;
    float* B3 = B2 + (size_t)N * N_HID;
    float* B4 = B3 + (size_t)N * N_OUT;

    const int TB = 256;
    const int mtiles = cdiv(N, 16);             // N = 100000 -> 6250 (exact)
    dim3 gemmBlk(128);
    dim3 gH(cdiv(mtiles, 4), N_HID / 16);       // (x, 4) col tiles for 64
    dim3 gO(cdiv(mtiles, 4), cdiv(N_OUT, 16));  // (x, 3) col tiles for 40

    // --- GCN normalization ---------------------------------------------------
    deg_init_kernel<<<cdiv(N, TB), TB, 0, stream>>>(dinv, N);
    deg_accum_kernel<<<cdiv(E, TB), TB, 0, stream>>>(dstIdx, dinv, E);
    dinv_kernel<<<cdiv(N, TB), TB, 0, stream>>>(dinv, N);

    // --- Layer 1: HW1 = x @ Wc1  (N x 128) @ (128 x 64) ----------------------
    wmma_gemm_f32<N_IN, N_HID, 0, N_IN, N_HID, N_HID>
        <<<gH, gemmBlk, 0, stream>>>(x, Wc1, B1, mtiles);
    agg_init_kernel<N_HID><<<cdiv(N * N_HID, TB), TB, 0, stream>>>(B1, dinv, bc1, B2, N);
    edge_scatter_kernel<N_HID><<<cdiv(E, TB), TB, 0, stream>>>(srcIdx, dstIdx, dinv,
                                                               B1, B2, E);
    relu_kernel<<<cdiv(N * N_HID, TB), TB, 0, stream>>>(B2, N * N_HID);

    // --- Deletion 1: HL = h @ Wd1^T ; h_del = mask ? HL + bd1 : h ------------
    wmma_gemm_f32<N_HID, N_HID, 1, N_HID, N_HID, N_HID>
        <<<gH, gemmBlk, 0, stream>>>(B2, Wd1, B1, mtiles);
    select_affected_kernel<N_HID><<<cdiv(N * N_HID, TB), TB, 0, stream>>>(B1, bd1, B2,
                                                                          mask, B1, N);

    // --- Layer 2: HW2 = h_del @ Wc2  (N x 64) @ (64 x 40) --------------------
    wmma_gemm_f32<N_HID, N_OUT, 0, N_HID, N_OUT, N_OUT>
        <<<gO, gemmBlk, 0, stream>>>(B1, Wc2, B3, mtiles);
    agg_init_kernel<N_OUT><<<cdiv(N * N_OUT, TB), TB, 0, stream>>>(B3, dinv, bc2, B4, N);
    edge_scatter_kernel<N_OUT><<<cdiv(E, TB), TB, 0, stream>>>(srcIdx, dstIdx, dinv,
                                                               B3, B4, E);

    // --- Deletion 2: OL = agg2 @ Wd2^T ; out = mask ? OL + bd2 : agg2 --------
    wmma_gemm_f32<N_OUT, N_OUT, 1, N_OUT, N_OUT, N_OUT>
        <<<gO, gemmBlk, 0, stream>>>(B4, Wd2, B3, mtiles);
    select_affected_kernel<N_OUT><<<cdiv(N * N_OUT, TB), TB, 0, stream>>>(B3, bd2, B4,
                                                                          mask,
                                                                          (float*)d_out, N);
}